// NNConvGNN_65910568125155
// MI455X (gfx1250) — compile-verified
//
#include <hip/hip_runtime.h>
#include <hip/hip_bf16.h>
#include <math.h>

#define N_NODES 50000
#define N_EDGES 500000
#define ULD 416      // padded u row length (26 WMMA n-tiles)
#define UVALID 408   // 384 weight cols + 24 bias cols

typedef __attribute__((ext_vector_type(2))) float v2f;
typedef __attribute__((ext_vector_type(8))) float v8f;

// ---------------------------------------------------------------- init ------
__global__ __launch_bounds__(256) void k_init(float* agg, float* deg, int clear_deg) {
    int t = blockIdx.x * blockDim.x + threadIdx.x;
    if (t < N_NODES * 24) {
        int col = t % 24;
        agg[t] = (col >= 8 && col < 16) ? -INFINITY : 0.0f;
    }
    if (clear_deg && t < N_NODES) deg[t] = 0.0f;
}

// ------------------------------------------------------------ node prep -----
// x16 = relu(relu(x_feat @ Wp + bp) @ Wt + bt)
__global__ __launch_bounds__(256) void k_prep(const float* __restrict__ xf,
                                              const float* __restrict__ Wp, const float* __restrict__ bp,
                                              const float* __restrict__ Wt, const float* __restrict__ bt,
                                              float* __restrict__ x16) {
    __shared__ float sWp[24 * 32], sbp[32], sWt[32 * 16], sbt[16];
    for (int i = threadIdx.x; i < 24 * 32; i += blockDim.x) sWp[i] = Wp[i];
    for (int i = threadIdx.x; i < 32 * 16; i += blockDim.x) sWt[i] = Wt[i];
    if (threadIdx.x < 32) sbp[threadIdx.x] = bp[threadIdx.x];
    if (threadIdx.x < 16) sbt[threadIdx.x] = bt[threadIdx.x];
    __syncthreads();
    int n = blockIdx.x * blockDim.x + threadIdx.x;
    if (n >= N_NODES) return;
    float x[24], h[32];
#pragma unroll
    for (int i = 0; i < 24; ++i) x[i] = xf[(size_t)n * 24 + i];
#pragma unroll
    for (int j = 0; j < 32; ++j) {
        float a = sbp[j];
#pragma unroll
        for (int i = 0; i < 24; ++i) a = fmaf(x[i], sWp[i * 32 + j], a);
        h[j] = fmaxf(a, 0.0f);
    }
#pragma unroll
    for (int j = 0; j < 16; ++j) {
        float a = sbt[j];
#pragma unroll
        for (int i = 0; i < 32; ++i) a = fmaf(h[i], sWt[i * 16 + j], a);
        x16[(size_t)n * 16 + j] = fmaxf(a, 0.0f);
    }
}

// ---------------------------------------------------- edge trunk + degree ---
__global__ __launch_bounds__(256) void k_trunk(const float* __restrict__ ea,
                                               const int* __restrict__ ei,
                                               const float* __restrict__ W1, const float* __restrict__ b1,
                                               const float* __restrict__ W2, const float* __restrict__ b2,
                                               float* __restrict__ erep, float* __restrict__ deg) {
    __shared__ float sW1[16 * 32], sb1[32], sW2[32 * 16], sb2[16];
    for (int i = threadIdx.x; i < 16 * 32; i += blockDim.x) sW1[i] = W1[i];
    for (int i = threadIdx.x; i < 32 * 16; i += blockDim.x) sW2[i] = W2[i];
    if (threadIdx.x < 32) sb1[threadIdx.x] = b1[threadIdx.x];
    if (threadIdx.x < 16) sb2[threadIdx.x] = b2[threadIdx.x];
    __syncthreads();
    int e = blockIdx.x * blockDim.x + threadIdx.x;
    if (e >= N_EDGES) return;
    float a[16], h[32];
#pragma unroll
    for (int i = 0; i < 16; ++i) a[i] = ea[(size_t)e * 16 + i];
#pragma unroll
    for (int j = 0; j < 32; ++j) {
        float s = sb1[j];
#pragma unroll
        for (int i = 0; i < 16; ++i) s = fmaf(a[i], sW1[i * 32 + j], s);
        h[j] = fmaxf(s, 0.0f);
    }
#pragma unroll
    for (int j = 0; j < 16; ++j) {
        float s = sb2[j];
#pragma unroll
        for (int i = 0; i < 32; ++i) s = fmaf(h[i], sW2[i * 16 + j], s);
        erep[(size_t)e * 16 + j] = 1.0f / (1.0f + __expf(-s));
    }
    int dst = ei[N_EDGES + e];
    atomicAdd(deg + dst, 1.0f);
}

// --------------------------------------------- u = X @ B  via f32 WMMA ------
// B[i, k*24 + a*8 + o] = eW[a][k][i*8+o];  B[i, 384 + a*8 + o] = eB[a][i*8+o]
template <int IN>
__global__ __launch_bounds__(256) void k_build_u(const float* __restrict__ X,
                                                 const float* __restrict__ eW,
                                                 const float* __restrict__ eB,
                                                 float* __restrict__ U, int mtiles) {
    constexpr int KST = IN / 4;
    __shared__ float Bs[IN * ULD];
    for (int i = threadIdx.x; i < IN * ULD; i += blockDim.x) {
        int irow = i / ULD, c = i % ULD;
        float v = 0.0f;
        if (c < 384) {
            int k = c / 24, rem = c % 24, a = rem >> 3, o = rem & 7;
            v = eW[((a * 16 + k) * IN + irow) * 8 + o];
        } else if (c < UVALID) {
            int t = c - 384, a = t >> 3, o = t & 7;
            v = eB[(a * IN + irow) * 8 + o];
        }
        Bs[i] = v;
    }
    __syncthreads();
    int lane = threadIdx.x & 31;
    int wave = threadIdx.x >> 5;
    int mt = blockIdx.x * 8 + wave;
    if (mt >= mtiles) return;
    int row = lane & 15, hi = lane >> 4;
    int base = mt * 16;
    v2f a[KST];
#pragma unroll
    for (int kk = 0; kk < KST; ++kk) {
        const float* xp = X + (size_t)(base + row) * IN + kk * 4 + 2 * hi;
        a[kk][0] = xp[0];
        a[kk][1] = xp[1];
    }
    for (int nn = 0; nn < 26; ++nn) {
        v8f c = {0.f, 0.f, 0.f, 0.f, 0.f, 0.f, 0.f, 0.f};
#pragma unroll
        for (int kk = 0; kk < KST; ++kk) {
            int kb = kk * 4 + 2 * hi;
            v2f b;
            b[0] = Bs[kb * ULD + nn * 16 + row];
            b[1] = Bs[(kb + 1) * ULD + nn * 16 + row];
            c = __builtin_amdgcn_wmma_f32_16x16x4_f32(false, a[kk], false, b,
                                                      (short)0, c, false, false);
        }
        int col = nn * 16 + row;
        if (col < UVALID) {
#pragma unroll
            for (int r = 0; r < 8; ++r)
                U[(size_t)(base + r + 8 * hi) * ULD + col] = c[r];
        }
    }
}

// ------------------------------------------------ edge messages + scatter ---
__device__ inline void atomicMaxF(float* addr, float v) {
    if (v >= 0.0f) atomicMax((int*)addr, __float_as_int(v));
    else           atomicMin((unsigned int*)addr, __float_as_uint(v));
}

__global__ __launch_bounds__(256) void k_msg(const int* __restrict__ ei,
                                             const float* __restrict__ erep,
                                             const float* __restrict__ U,
                                             float* __restrict__ agg) {
    int e = blockIdx.x * blockDim.x + threadIdx.x;
    if (e >= N_EDGES) return;
    int src = ei[e], dst = ei[N_EDGES + e];
    float er[16];
    const float4* ep = (const float4*)(erep + (size_t)e * 16);
#pragma unroll
    for (int q = 0; q < 4; ++q) {
        float4 v = ep[q];
        er[q * 4 + 0] = v.x; er[q * 4 + 1] = v.y; er[q * 4 + 2] = v.z; er[q * 4 + 3] = v.w;
    }
    const float* urow = U + (size_t)src * ULD;
    float acc[24];
#pragma unroll
    for (int q = 0; q < 6; ++q) {
        float4 v = ((const float4*)(urow + 384))[q];
        acc[q * 4 + 0] = v.x; acc[q * 4 + 1] = v.y; acc[q * 4 + 2] = v.z; acc[q * 4 + 3] = v.w;
    }
#pragma unroll 4
    for (int k = 0; k < 16; ++k) {
        float f = er[k];
        const float4* up = (const float4*)(urow + k * 24);
#pragma unroll
        for (int q = 0; q < 6; ++q) {
            float4 v = up[q];
            acc[q * 4 + 0] = fmaf(f, v.x, acc[q * 4 + 0]);
            acc[q * 4 + 1] = fmaf(f, v.y, acc[q * 4 + 1]);
            acc[q * 4 + 2] = fmaf(f, v.z, acc[q * 4 + 2]);
            acc[q * 4 + 3] = fmaf(f, v.w, acc[q * 4 + 3]);
        }
    }
    float* ag = agg + (size_t)dst * 24;
#pragma unroll
    for (int o = 0; o < 8; ++o)  atomicAdd(ag + o, acc[o]);
#pragma unroll
    for (int o = 8; o < 16; ++o) atomicMaxF(ag + o, acc[o]);
#pragma unroll
    for (int o = 16; o < 24; ++o) atomicAdd(ag + o, acc[o]);
}

// --------------------------------------------------------- finalize conv ----
template <int IN>
__global__ __launch_bounds__(256) void k_finalize(const float* __restrict__ X,
                                                  const float* __restrict__ agg,
                                                  const float* __restrict__ deg,
                                                  const float* __restrict__ rW,
                                                  const float* __restrict__ bias,
                                                  float* __restrict__ Y) {
    __shared__ float srW[3 * IN * 8], sb[24];
    for (int i = threadIdx.x; i < 3 * IN * 8; i += blockDim.x) srW[i] = rW[i];
    if (threadIdx.x < 24) sb[threadIdx.x] = bias[threadIdx.x];
    __syncthreads();
    int n = blockIdx.x * blockDim.x + threadIdx.x;
    if (n >= N_NODES) return;
    float x[IN];
#pragma unroll
    for (int i = 0; i < IN; ++i) x[i] = X[(size_t)n * IN + i];
    float d = deg[n];
    float dm = fmaxf(d, 1.0f);
    bool has = d > 0.5f;
#pragma unroll
    for (int a = 0; a < 3; ++a) {
#pragma unroll
        for (int o = 0; o < 8; ++o) {
            float r = sb[a * 8 + o];
#pragma unroll
            for (int i = 0; i < IN; ++i) r = fmaf(x[i], srW[(a * IN + i) * 8 + o], r);
            float g = agg[(size_t)n * 24 + a * 8 + o];
            float val;
            if (a == 0)      val = g / dm + r;            // mean
            else if (a == 1) val = (has ? g : 0.0f) + r;  // max
            else             val = g + r;                 // add
            Y[(size_t)n * 24 + a * 8 + o] = fmaxf(val, 0.0f);
        }
    }
}

// ------------------------------------------------------------- head ---------
__global__ __launch_bounds__(256) void k_head(const float* __restrict__ X,
                                              const float* __restrict__ W,
                                              const float* __restrict__ b,
                                              float* __restrict__ out) {
    int n = blockIdx.x * blockDim.x + threadIdx.x;
    if (n >= N_NODES) return;
    float l0 = b[0], l1 = b[1];
#pragma unroll
    for (int i = 0; i < 24; ++i) {
        float x = X[(size_t)n * 24 + i];
        l0 = fmaf(x, W[i * 2 + 0], l0);
        l1 = fmaf(x, W[i * 2 + 1], l1);
    }
    float m = fmaxf(l0, l1);
    float e0 = __expf(l0 - m), e1 = __expf(l1 - m);
    float inv = 1.0f / (e0 + e1);
    out[(size_t)n * 2 + 0] = l0;
    out[(size_t)n * 2 + 1] = l1;
    out[(size_t)N_NODES * 2 + (size_t)n * 2 + 0] = e0 * inv;
    out[(size_t)N_NODES * 2 + (size_t)n * 2 + 1] = e1 * inv;
}

// ------------------------------------------------------------- launch -------
extern "C" void kernel_launch(void* const* d_in, const int* in_sizes, int n_in,
                              void* d_out, int out_size, void* d_ws, size_t ws_size,
                              hipStream_t stream) {
    (void)in_sizes; (void)n_in; (void)out_size; (void)ws_size;
    const float* x_feat = (const float*)d_in[0];
    const int*   ei     = (const int*)d_in[1];
    const float* eattr  = (const float*)d_in[2];
    const float* Wp = (const float*)d_in[3];  const float* bp = (const float*)d_in[4];
    const float* Wt = (const float*)d_in[5];  const float* bt = (const float*)d_in[6];
    const float* eW1 = (const float*)d_in[7]; const float* eb1 = (const float*)d_in[8];
    const float* eW2 = (const float*)d_in[9]; const float* eb2 = (const float*)d_in[10];
    const float* c1eW = (const float*)d_in[11]; const float* c1eB = (const float*)d_in[12];
    const float* c1rW = (const float*)d_in[13]; const float* c1b  = (const float*)d_in[14];
    const float* c2eW = (const float*)d_in[15]; const float* c2eB = (const float*)d_in[16];
    const float* c2rW = (const float*)d_in[17]; const float* c2b  = (const float*)d_in[18];
    const float* oW = (const float*)d_in[19]; const float* ob = (const float*)d_in[20];
    float* out = (float*)d_out;

    char* ws = (char*)d_ws;
    size_t off = 0;
    auto take = [&](size_t bytes) -> char* {
        char* p = ws + off;
        off += (bytes + 255) & ~(size_t)255;
        return p;
    };
    float* x16  = (float*)take((size_t)N_NODES * 16 * 4);
    float* erep = (float*)take((size_t)N_EDGES * 16 * 4);
    float* deg  = (float*)take((size_t)N_NODES * 4);
    float* U    = (float*)take((size_t)N_NODES * ULD * 4);
    float* agg  = (float*)take((size_t)N_NODES * 24 * 4);
    float* x24a = (float*)take((size_t)N_NODES * 24 * 4);
    float* x24b = (float*)take((size_t)N_NODES * 24 * 4);

    const int B = 256;
    const int gN  = (N_NODES + B - 1) / B;
    const int gE  = (N_EDGES + B - 1) / B;
    const int gI  = (N_NODES * 24 + B - 1) / B;
    const int mtiles = N_NODES / 16;              // 3125
    const int gU  = (mtiles + 7) / 8;             // 8 waves / block, 1 m-tile / wave

    k_init<<<gI, B, 0, stream>>>(agg, deg, 1);
    k_prep<<<gN, B, 0, stream>>>(x_feat, Wp, bp, Wt, bt, x16);
    k_trunk<<<gE, B, 0, stream>>>(eattr, ei, eW1, eb1, eW2, eb2, erep, deg);

    // ---- conv1 ----
    k_build_u<16><<<gU, B, 0, stream>>>(x16, c1eW, c1eB, U, mtiles);
    k_msg<<<gE, B, 0, stream>>>(ei, erep, U, agg);
    k_finalize<16><<<gN, B, 0, stream>>>(x16, agg, deg, c1rW, c1b, x24a);

    // ---- conv2 ----
    k_init<<<gI, B, 0, stream>>>(agg, deg, 0);
    k_build_u<24><<<gU, B, 0, stream>>>(x24a, c2eW, c2eB, U, mtiles);
    k_msg<<<gE, B, 0, stream>>>(ei, erep, U, agg);
    k_finalize<24><<<gN, B, 0, stream>>>(x24a, agg, deg, c2rW, c2b, x24b);

    k_head<<<gN, B, 0, stream>>>(x24b, oW, ob, out);
}